// MEPG_Loss_13357348290561
// MI455X (gfx1250) — compile-verified
//
#include <hip/hip_runtime.h>
#include <hip/hip_bf16.h>
#include <math.h>

typedef __attribute__((ext_vector_type(2))) float v2f;
typedef __attribute__((ext_vector_type(8))) float v8f;

#define N_S     2048
#define T_LEN   512
#define N_HID   128
#define N_ACT   4
#define SD_F    0.04f
#define ALPHA_F 0.1f

// ---------------------------------------------------------------------------
// Fast tanh: native V_TANH_F32 on gfx1250 if the builtin exists, else a
// branchless exp2-based evaluation (no EXEC divergence, ~6 VALU ops).
// ---------------------------------------------------------------------------
__device__ __forceinline__ float fast_tanh(float x) {
#if __has_builtin(__builtin_amdgcn_tanhf)
    return __builtin_amdgcn_tanhf(x);
#else
    const float a = __builtin_fabsf(x);
    // e = exp(-2a) = exp2(-2a * log2(e)); a >= 0 so e in (0, 1], no overflow
    const float e = __builtin_amdgcn_exp2f(a * -2.8853900817779268f);
    const float t = (1.0f - e) * __builtin_amdgcn_rcpf(1.0f + e);
    return __builtin_copysignf(t, x);
#endif
}

// ---------------------------------------------------------------------------
// Pass 1: one wave per 16-row (s, t0..t0+15) tile.
//   GEMM1 (16x16 @ 16x128) and GEMM2 (16x128 @ 128x4, N zero-padded to 16)
//   via V_WMMA_F32_16X16X4_F32. Emits per-tile sum(ll) and ll at t = T-1.
// ---------------------------------------------------------------------------
__global__ __launch_bounds__(128) void mepg_pass1(
    const float* __restrict__ states,   // [S,16,T]
    const float* __restrict__ actions,  // [S,4,T]
    const float* __restrict__ W1,       // [16,128]
    const float* __restrict__ b1,       // [128]
    const float* __restrict__ W2,       // [128,4]
    const float* __restrict__ b2,       // [4]
    float* __restrict__ partial,        // [S*T/16] per-tile sum of ll
    float* __restrict__ llLast)         // [S]      ll at t=T-1
{
    // LDS staging for the H (16x128) tile per wave; stride 130 floats so
    // column reads (stride*row) hit distinct banks. 4 waves/block -> ~32.5 KB.
    __shared__ float hbuf[4][16 * 130];

    const int lane = threadIdx.x & 31;
    const int wave = threadIdx.x >> 5;
    const int tile = blockIdx.x * 4 + wave;    // 0 .. 65535
    const int sIdx = tile >> 5;                // 32 tiles per sim (T/16)
    const int t0   = (tile & 31) << 4;
    const int m    = lane & 15;                // row (A) / col (B,C) index
    const int kh   = lane >> 4;                // K half select

    // ---- A-matrix: state tile, WMMA f32 16x4 layout, 4 K-chunks ----------
    // X[row][d] = states[sIdx*16*T + d*T + (t0+row)]
    const float* sbase = states + (size_t)sIdx * (16 * T_LEN) + t0 + m;
    v2f a[4];
#pragma unroll
    for (int c = 0; c < 4; ++c) {
        const int k0 = 4 * c + 2 * kh;
        a[c].x = sbase[(size_t)k0 * T_LEN];
        a[c].y = sbase[(size_t)(k0 + 1) * T_LEN];
    }

    float* hrow = &hbuf[wave][0];

    // ---- GEMM1: H = tanh(X @ W1 + b1), 8 N-tiles x 4 K-chunks ------------
#pragma unroll
    for (int n = 0; n < 8; ++n) {
        const float b1v = b1[n * 16 + m];
        v8f acc = { b1v, b1v, b1v, b1v, b1v, b1v, b1v, b1v };
#pragma unroll
        for (int c = 0; c < 4; ++c) {
            const int k0 = 4 * c + 2 * kh;
            v2f b;
            b.x = W1[k0 * N_HID + n * 16 + m];
            b.y = W1[(k0 + 1) * N_HID + n * 16 + m];
            acc = __builtin_amdgcn_wmma_f32_16x16x4_f32(
                false, a[c], false, b, (short)0, acc, false, false);
        }
        // tanh, stash to LDS row-major (row = kh*8+g, col = n*16+m)
#pragma unroll
        for (int g = 0; g < 8; ++g) {
            const int row = kh * 8 + g;
            hrow[row * 130 + n * 16 + m] = fast_tanh(acc[g]);
        }
    }

    // ---- GEMM2: C = (mu - act) = H @ W2pad + (b2 - act), cols 0..3 valid --
    v8f acc2;
#pragma unroll
    for (int g = 0; g < 8; ++g) {
        const int row = kh * 8 + g;
        float v = 0.0f;
        if (m < N_ACT)
            v = b2[m] - actions[(size_t)sIdx * (N_ACT * T_LEN) + m * T_LEN + t0 + row];
        acc2[g] = v;
    }
#pragma unroll
    for (int c = 0; c < 32; ++c) {
        const int k0 = 4 * c + 2 * kh;
        const v2f a2 = *(const v2f*)&hrow[m * 130 + k0];   // ds_load_b64
        v2f bm;
        bm.x = (m < N_ACT) ? W2[k0 * N_ACT + m] : 0.0f;
        bm.y = (m < N_ACT) ? W2[(k0 + 1) * N_ACT + m] : 0.0f;
        acc2 = __builtin_amdgcn_wmma_f32_16x16x4_f32(
            false, a2, false, bm, (short)0, acc2, false, false);
    }

    // ---- log-lik per row: padded cols are exactly 0 -> xor-reduce halves --
    const float LLC = -2.0f * logf(6.2831853071795864f * SD_F); // -0.5*d*log(2*pi*SD)
    float ll[8];
#pragma unroll
    for (int g = 0; g < 8; ++g) {
        float sq = acc2[g] * acc2[g];
        sq += __shfl_xor(sq, 1, 32);
        sq += __shfl_xor(sq, 2, 32);
        sq += __shfl_xor(sq, 4, 32);
        sq += __shfl_xor(sq, 8, 32);             // sum over j within 16-lane half
        ll[g] = -0.5f * sq * (1.0f / SD_F) + LLC;
    }

    float tl = 0.0f;
#pragma unroll
    for (int g = 0; g < 8; ++g) tl += ll[g];     // 8 rows of this half
    tl += __shfl_xor(tl, 16, 32);                // + other half -> 16-row sum

    if (lane == 0) partial[tile] = tl;
    // t = T-1 lives in the last tile of each sim, row 15 (half 1, g = 7)
    if (t0 == (T_LEN - 16) && lane == 16) llLast[sIdx] = ll[7];
}

// ---------------------------------------------------------------------------
// Pass 2 (single block): per-sim analytic fold of A_hat, final scalar.
//   sumA[s] = sum_t rewards - a*sumL + (rew[T-1] - a*ll[T-1]) - T*log(0.5)
//   out = sum_s sumA[s]*sumL[s] / S
// ---------------------------------------------------------------------------
__global__ __launch_bounds__(256) void mepg_pass2(
    const float* __restrict__ rewards,   // [S,T]
    const float* __restrict__ partial,   // [S*32]
    const float* __restrict__ llLast,    // [S]
    float* __restrict__ out)
{
    __shared__ float wsum[8];
    const int lane = threadIdx.x & 31;
    const int wave = threadIdx.x >> 5;
    const float mx_pos = logf(0.5f);

    float contrib = 0.0f;
    for (int s = wave; s < N_S; s += 8) {
        float r = 0.0f;
        for (int t = lane; t < T_LEN; t += 32) r += rewards[s * T_LEN + t];
        float p = partial[s * 32 + lane];          // exactly 32 tiles per sim
#pragma unroll
        for (int off = 16; off; off >>= 1) {
            r += __shfl_xor(r, off, 32);
            p += __shfl_xor(p, off, 32);
        }
        if (lane == 0) {
            const float sumL     = p;
            const float baseLast = rewards[s * T_LEN + (T_LEN - 1)] - ALPHA_F * llLast[s];
            const float sumA     = (r - ALPHA_F * sumL) + baseLast - (float)T_LEN * mx_pos;
            contrib += sumA * sumL;
        }
    }
    if (lane == 0) wsum[wave] = contrib;
    __syncthreads();
    if (threadIdx.x == 0) {
        float tot = 0.0f;
        for (int w = 0; w < 8; ++w) tot += wsum[w];
        out[0] = tot / (float)N_S;
    }
}

// ---------------------------------------------------------------------------
extern "C" void kernel_launch(void* const* d_in, const int* in_sizes, int n_in,
                              void* d_out, int out_size, void* d_ws, size_t ws_size,
                              hipStream_t stream) {
    const float* states  = (const float*)d_in[0];
    const float* actions = (const float*)d_in[1];
    const float* rewards = (const float*)d_in[2];
    const float* W1      = (const float*)d_in[3];
    const float* b1      = (const float*)d_in[4];
    const float* W2      = (const float*)d_in[5];
    const float* b2      = (const float*)d_in[6];
    float* out = (float*)d_out;

    float* partial = (float*)d_ws;                   // 65536 floats
    float* llLast  = partial + (N_S * T_LEN / 16);   // 2048 floats

    const int tiles  = N_S * T_LEN / 16;             // 65536
    const int blocks = tiles / 4;                    // 4 waves (128 thr) / block

    mepg_pass1<<<blocks, 128, 0, stream>>>(states, actions, W1, b1, W2, b2,
                                           partial, llLast);
    mepg_pass2<<<1, 256, 0, stream>>>(rewards, partial, llLast, out);
}